// EncoderRNN_764504178844
// MI455X (gfx1250) — compile-verified
//
#include <hip/hip_runtime.h>
#include <hip/hip_bf16.h>
#include <math.h>

#define HID   2048
#define G3H   6144          // 3*HID
#define ROWS  12288         // 2*3*HID (gi rows then gh rows)

typedef __attribute__((ext_vector_type(2))) float v2f;
typedef __attribute__((ext_vector_type(8))) float v8f;

// One wave computes a 16-row x kChunk partial of W*vec with V_WMMA_F32_16X16X4_F32.
// Block = 128 threads = 4 waves = 64 rows. Grid = 192 * splitK blocks.
// blockIdx.x / splitK selects the 64-row group (0..95 -> w_ih*x, 96..191 -> w_hh*h)
// blockIdx.x % splitK selects the K chunk; partial sums go to g[part*ROWS + row].
__global__ __launch_bounds__(128) void gru_gemv_wmma(
    const int*   __restrict__ tok,
    const float* __restrict__ hidden,   // [HID]
    const float* __restrict__ emb,      // [VOCAB, HID]
    const float* __restrict__ w_ih,     // [3H, H]
    const float* __restrict__ w_hh,     // [3H, H]
    float*       __restrict__ g,        // [splitK, ROWS] partials
    int kChunk, int splitK)
{
    __shared__ float xs[HID];

    const int part    = blockIdx.x % splitK;
    const int rowBase = (blockIdx.x / splitK) * 64;       // global row of block
    const int kBase   = part * kChunk;
    const bool isIH   = rowBase < G3H;

    // Stage this block's K-chunk of the operand vector into LDS (b128 loads).
    const float* vec = (isIH ? (emb + (size_t)tok[0] * HID) : hidden) + kBase;
    for (int i = threadIdx.x * 4; i < kChunk; i += 128 * 4)
        *(float4*)(xs + i) = *(const float4*)(vec + i);
    __syncthreads();

    const int wave = threadIdx.x >> 5;
    const int lane = threadIdx.x & 31;
    const int rowTileBase = rowBase + wave * 16;          // global row of D row 0
    const int localBase   = isIH ? rowTileBase : rowTileBase - G3H;
    const float* __restrict__ W = isIH ? w_ih : w_hh;

    // A layout (16x4 f32): lane<16 -> rows, K={0,1}; lane>=16 -> K={2,3}.
    // B layout (4x16 f32): identical k + 2*(lane>>4) pattern with cols broadcast.
    const int kOff    = (lane >> 4) * 2;
    const int laneRow = localBase + (lane & 15);
    const float* aPtr = W + (size_t)laneRow * HID + kBase + kOff;
    const float* bPtr = xs + kOff;

    v8f acc0 = {0.f, 0.f, 0.f, 0.f, 0.f, 0.f, 0.f, 0.f};
    v8f acc1 = {0.f, 0.f, 0.f, 0.f, 0.f, 0.f, 0.f, 0.f};

    // Two independent accumulators break the serial WMMA D->C chain; deep
    // unroll keeps ~16 global b64 loads in flight per wave.
    #pragma unroll 8
    for (int k = 0; k < kChunk; k += 8) {
        v2f a0 = *(const v2f*)(aPtr + k);        // global_load_b64
        v2f b0 = *(const v2f*)(bPtr + k);        // ds_load_b64 (broadcast)
        v2f a1 = *(const v2f*)(aPtr + k + 4);
        v2f b1 = *(const v2f*)(bPtr + k + 4);
        acc0 = __builtin_amdgcn_wmma_f32_16x16x4_f32(
            false, a0, false, b0, (short)0, acc0, false, false);
        acc1 = __builtin_amdgcn_wmma_f32_16x16x4_f32(
            false, a1, false, b1, (short)0, acc1, false, false);
    }
    v8f acc = acc0 + acc1;

    // D layout: lane 0 holds rows 0..7 (N=0), lane 16 holds rows 8..15.
    if ((lane & 15) == 0) {
        const int mbase = part * ROWS + rowTileBase + ((lane >> 4) ? 8 : 0);
        #pragma unroll
        for (int v = 0; v < 8; ++v)
            g[mbase + v] = acc[v];
    }
}

// Fused GRU gate math; sums split-K partials, writes (out, h_new) == h_new twice.
__global__ __launch_bounds__(256) void gru_gates(
    const float* __restrict__ g,        // [splitK, ROWS]
    const float* __restrict__ hidden,
    const float* __restrict__ b_ih,
    const float* __restrict__ b_hh,
    float*       __restrict__ out,      // [2*HID]
    int splitK)
{
    const int i = blockIdx.x * 256 + threadIdx.x;
    if (i >= HID) return;

    float i_r = b_ih[i], i_z = b_ih[i + HID], i_n = b_ih[i + 2 * HID];
    float h_r = b_hh[i], h_z = b_hh[i + HID], h_n = b_hh[i + 2 * HID];
    for (int p = 0; p < splitK; ++p) {
        const float* gp = g + (size_t)p * ROWS;
        i_r += gp[i];
        i_z += gp[i + HID];
        i_n += gp[i + 2 * HID];
        h_r += gp[G3H + i];
        h_z += gp[G3H + i + HID];
        h_n += gp[G3H + i + 2 * HID];
    }

    const float r = 1.f / (1.f + expf(-(i_r + h_r)));
    const float z = 1.f / (1.f + expf(-(i_z + h_z)));
    const float n = tanhf(i_n + r * h_n);
    const float h = hidden[i];
    const float hn = (1.f - z) * n + z * h;

    out[i]       = hn;   // output  [1,1,H]
    out[HID + i] = hn;   // hidden' [1,1,H]
}

extern "C" void kernel_launch(void* const* d_in, const int* in_sizes, int n_in,
                              void* d_out, int out_size, void* d_ws, size_t ws_size,
                              hipStream_t stream) {
    const int*   tok    = (const int*)  d_in[0];
    const float* hidden = (const float*)d_in[1];
    const float* emb    = (const float*)d_in[2];
    const float* w_ih   = (const float*)d_in[3];
    const float* w_hh   = (const float*)d_in[4];
    const float* b_ih   = (const float*)d_in[5];
    const float* b_hh   = (const float*)d_in[6];
    float* out = (float*)d_out;
    float* g   = (float*)d_ws;

    // Split-K factor chosen by available scratch (4 -> 192 KB of partials).
    int splitK = 1;
    if (ws_size >= 4 * (size_t)ROWS * sizeof(float))      splitK = 4;
    else if (ws_size >= 2 * (size_t)ROWS * sizeof(float)) splitK = 2;
    const int kChunk = HID / splitK;

    gru_gemv_wmma<<<192 * splitK, 128, 0, stream>>>(
        tok, hidden, emb, w_ih, w_hh, g, kChunk, splitK);
    gru_gates<<<(HID + 255) / 256, 256, 0, stream>>>(
        g, hidden, b_ih, b_hh, out, splitK);
}